// SAGE_DIFFPOOL_2370821948030
// MI455X (gfx1250) — compile-verified
//
#include <hip/hip_runtime.h>
#include <hip/hip_bf16.h>
#include <math.h>

// ---------------- problem constants ----------------
#define BGR   16
#define NN    2048
#define NODES (BGR*NN)      // 32768
#define EDG   (BGR*65536)   // 1048576
#define NC    100
#define EPSBN 1e-5f

typedef float v2f __attribute__((ext_vector_type(2)));
typedef float v8f __attribute__((ext_vector_type(8)));

static __device__ __forceinline__ void atomAdd(float* p, float v) {
    __hip_atomic_fetch_add(p, v, __ATOMIC_RELAXED, __HIP_MEMORY_SCOPE_AGENT);
}

// ---------------- tiny utility kernels ----------------
__global__ void k_zero(float* p, long n) {
    long i = (long)blockIdx.x * blockDim.x + threadIdx.x;
    if (i < n) p[i] = 0.0f;
}
__global__ void k_fill(float* p, float v, long n) {
    long i = (long)blockIdx.x * blockDim.x + threadIdx.x;
    if (i < n) p[i] = v;
}

// degrees: deg[c] += w (and +1 unweighted) per edge; buffers pre-filled with 1.0 (self loop)
__global__ void k_edge_deg(const int* __restrict__ col, const float* __restrict__ w,
                           float* degw, float* deg1, int E) {
    int e = blockIdx.x * blockDim.x + threadIdx.x;
    if (e >= E) return;
    int c = col[e];
    atomAdd(&degw[c], w[e]);
    atomAdd(&deg1[c], 1.0f);
}
__global__ void k_dinv(float* d, long n) {
    long i = (long)blockIdx.x * blockDim.x + threadIdx.x;
    if (i >= n) return;
    float v = d[i];
    d[i] = (v > 0.0f) ? rsqrtf(fmaxf(v, 1e-12f)) : 0.0f;
}

// ---------------- WMMA f32 GEMM (batched, optional transA) ----------------
// C[b] = A[b] x B[b];  A: MxK (row-major; or KxM accessed transposed), B: KxN, C: MxN.
// One 16x16 output tile per wave32 via V_WMMA_F32_16X16X4_F32.
//
// Branch-free main loop: lanes whose m-row / n-col fall outside the matrix load
// from a CLAMPED valid address instead of guarding with exec. Their garbage only
// reaches C rows >= M / cols >= N, which are never stored. Only the K remainder
// needs zero-fill (guarded epilogue, uniform branch + cndmask selects).
__global__ void k_gemm_wmma(const float* __restrict__ A, const float* __restrict__ B,
                            float* __restrict__ C,
                            int M, int N, int K, int lda, int ldb, int ldc,
                            long sA, long sB, long sC, int transA) {
    const float* Ab = A + (long)blockIdx.y * sA;
    const float* Bb = B + (long)blockIdx.y * sB;
    float*       Cb = C + (long)blockIdx.y * sC;

    int tM = (M + 15) >> 4, tN = (N + 15) >> 4;
    int tiles = tM * tN;
    int tile = blockIdx.x * (blockDim.x >> 5) + (threadIdx.x >> 5);
    if (tile >= tiles) return;
    int tm = tile / tN, tn = tile % tN;

    int lane = threadIdx.x & 31;
    int half = lane >> 4;       // 0: lanes 0-15, 1: lanes 16-31
    int lm   = lane & 15;
    int mrow = tm * 16 + lm;    // logical A row for this lane
    int ncol = tn * 16 + lm;    // logical B/C col for this lane
    int mc = min(mrow, M - 1);  // clamped (always-valid) indices
    int nc = min(ncol, N - 1);

    // per-lane base pointers; lane holds K = kk + 2*half + j (j = 0,1)
    long aJ = transA ? (long)lda : 1;                // stride between j=0 and j=1
    long aK = transA ? (long)lda * 4 : 4;            // stride per kk += 4
    const float* pa = Ab + (transA ? ((long)(2 * half) * lda + mc)
                                   : ((long)mc * lda + 2 * half));
    const float* pb = Bb + (long)(2 * half) * ldb + nc;
    long bK = (long)ldb * 4;

    v8f acc = {0.f,0.f,0.f,0.f,0.f,0.f,0.f,0.f};
    int Kfull = K & ~3;
#pragma unroll 4
    for (int kk = 0; kk < Kfull; kk += 4) {
        v2f a, b;
        a[0] = pa[0]; a[1] = pa[aJ];
        b[0] = pb[0]; b[1] = pb[ldb];
        pa += aK; pb += bK;
        acc = __builtin_amdgcn_wmma_f32_16x16x4_f32(false, a, false, b,
                                                    (short)0, acc, false, false);
    }
    if (Kfull < K) {            // K remainder: zero-fill k >= K via selects
        v2f a, b;
#pragma unroll
        for (int j = 0; j < 2; ++j) {
            int ka = Kfull + 2 * half + j;
            int kc = min(ka, K - 1);
            float av = transA ? Ab[(long)kc * lda + mc] : Ab[(long)mc * lda + kc];
            float bv = Bb[(long)kc * ldb + nc];
            a[j] = (ka < K) ? av : 0.f;
            b[j] = (ka < K) ? bv : 0.f;
        }
        acc = __builtin_amdgcn_wmma_f32_16x16x4_f32(false, a, false, b,
                                                    (short)0, acc, false, false);
    }
#pragma unroll
    for (int v = 0; v < 8; ++v) {
        int r = tm * 16 + v + 8 * half;   // C layout: VGPR v -> row v (+8 hi half)
        if (r < M && ncol < N) Cb[(long)r * ldc + ncol] = acc[v];
    }
}

// ---------------- sparse GCN pieces ----------------
// agg[i,f] = dinv[i]^2 * xw[i,f] + bias[f]   (self-loop weight 1, plus bias)
__global__ void k_selfloop_bias(const float* __restrict__ xw, const float* __restrict__ dinv,
                                const float* __restrict__ bias, float* __restrict__ agg,
                                long n, int C) {
    long i = (long)blockIdx.x * blockDim.x + threadIdx.x;
    if (i >= n * C) return;
    long r = i / C; int c = (int)(i % C);
    float d = dinv[r];
    agg[i] = d * d * xw[i] + bias[c];
}

// wave-per-edge, grid-stride: agg[col] += dinv[row]*w*dinv[col] * xw[row]
// 32 lanes hit consecutive addresses of one row -> coalesced L2 atomics
__global__ void k_edge_scatter(const int* __restrict__ row, const int* __restrict__ col,
                               const float* __restrict__ w, const float* __restrict__ dinv,
                               const float* __restrict__ xw, float* __restrict__ agg,
                               int E, int C) {
    int lane = threadIdx.x & 31;
    int wave = blockIdx.x * (blockDim.x >> 5) + (threadIdx.x >> 5);
    int nw   = gridDim.x * (blockDim.x >> 5);
    for (int e = wave; e < E; e += nw) {
        int r = row[e], c = col[e];
        float nrm = dinv[r] * dinv[c] * (w ? w[e] : 1.0f);
        const float* xr = xw + (long)r * C;
        float*       ac = agg + (long)c * C;
        for (int f = lane; f < C; f += 32) atomAdd(&ac[f], nrm * xr[f]);
    }
}

// ---------------- batch norm ----------------
__global__ void k_bn_stats(const float* __restrict__ X, int rows, int ld,
                           float* mean, float* var) {
    int c = blockIdx.x;
    __shared__ float ss[256], sq[256];
    float s = 0.f, q = 0.f;
    for (int i = threadIdx.x; i < rows; i += blockDim.x) {
        float v = X[(long)i * ld + c];
        s += v; q += v * v;
    }
    ss[threadIdx.x] = s; sq[threadIdx.x] = q; __syncthreads();
    for (int st = 128; st > 0; st >>= 1) {
        if (threadIdx.x < st) { ss[threadIdx.x] += ss[threadIdx.x + st];
                                sq[threadIdx.x] += sq[threadIdx.x + st]; }
        __syncthreads();
    }
    if (threadIdx.x == 0) {
        float m = ss[0] / rows;
        mean[c] = m;
        var[c]  = sq[0] / rows - m * m;   // biased variance
    }
}
__global__ void k_bn_apply(const float* __restrict__ X, long rows, int ldin,
                           const float* __restrict__ mean, const float* __restrict__ var,
                           const float* __restrict__ g, const float* __restrict__ beta,
                           float* __restrict__ Y, int ldout, int C) {
    long i = (long)blockIdx.x * blockDim.x + threadIdx.x;
    if (i >= rows * C) return;
    long r = i / C; int c = (int)(i % C);
    float v = (X[r * (long)ldin + c] - mean[c]) * rsqrtf(var[c] + EPSBN) * g[c] + beta[c];
    Y[r * (long)ldout + c] = v;
}

// per-graph max over nodes, written into out[b*outld + off + c]
__global__ void k_colmax(const float* __restrict__ X, int npg, int ld,
                         float* out, int outld, int off) {
    int c = blockIdx.x, b = blockIdx.y;
    __shared__ float sm[256];
    float m = -INFINITY;
    for (int i = threadIdx.x; i < npg; i += blockDim.x)
        m = fmaxf(m, X[((long)b * npg + i) * ld + c]);
    sm[threadIdx.x] = m; __syncthreads();
    for (int st = 128; st > 0; st >>= 1) {
        if (threadIdx.x < st) sm[threadIdx.x] = fmaxf(sm[threadIdx.x], sm[threadIdx.x + st]);
        __syncthreads();
    }
    if (threadIdx.x == 0) out[b * outld + off + c] = sm[0];
}

__global__ void k_add_bias(float* __restrict__ X, const float* __restrict__ b,
                           long rows, int C) {
    long i = (long)blockIdx.x * blockDim.x + threadIdx.x;
    if (i >= rows * C) return;
    X[i] += b[i % C];
}

// ---------------- softmax + entropy loss (wave per row) ----------------
__global__ void k_softmax_entropy(const float* __restrict__ s, float* __restrict__ p,
                                  float* loss, int rows, int C) {
    int lane = threadIdx.x & 31;
    int r = blockIdx.x * (blockDim.x >> 5) + (threadIdx.x >> 5);
    float ent = 0.f;
    if (r < rows) {
        const float* sr = s + (long)r * C;
        float*       pr = p + (long)r * C;
        float mx = -INFINITY;
        for (int f = lane; f < C; f += 32) mx = fmaxf(mx, sr[f]);
        for (int i = 16; i; i >>= 1) mx = fmaxf(mx, __shfl_xor(mx, i, 32));
        float sum = 0.f;
        for (int f = lane; f < C; f += 32) { float e = expf(sr[f] - mx); pr[f] = e; sum += e; }
        for (int i = 16; i; i >>= 1) sum += __shfl_xor(sum, i, 32);
        float inv = 1.f / sum;
        for (int f = lane; f < C; f += 32) {
            float v = pr[f] * inv; pr[f] = v;
            ent += -v * logf(v + 1e-15f);
        }
    }
    __shared__ float sm[256];
    sm[threadIdx.x] = ent; __syncthreads();
    for (int st = 128; st > 0; st >>= 1) {
        if (threadIdx.x < st) sm[threadIdx.x] += sm[threadIdx.x + st];
        __syncthreads();
    }
    if (threadIdx.x == 0) atomAdd(loss, sm[0] / rows);
}

// modularity: loss += -(1/E) * sum_e dot(p[row],p[col]); wave per edge, grid-stride
__global__ void k_modularity(const int* __restrict__ row, const int* __restrict__ col,
                             const float* __restrict__ p, float* loss, int E, int C) {
    int lane = threadIdx.x & 31;
    int wave = blockIdx.x * (blockDim.x >> 5) + (threadIdx.x >> 5);
    int nw   = gridDim.x * (blockDim.x >> 5);
    float acc = 0.f;                 // lane-partial of the global sum
    for (int e = wave; e < E; e += nw) {
        const float* pr = p + (long)row[e] * C;
        const float* pc = p + (long)col[e] * C;
        for (int f = lane; f < C; f += 32) acc += pr[f] * pc[f];
    }
    __shared__ float sm[256];
    sm[threadIdx.x] = acc; __syncthreads();
    for (int st = 128; st > 0; st >>= 1) {
        if (threadIdx.x < st) sm[threadIdx.x] += sm[threadIdx.x + st];
        __syncthreads();
    }
    if (threadIdx.x == 0) atomAdd(loss, -sm[0] / (float)E);
}

// T[row] += w * S[col]  (T = A*S per-edge scatter); wave per edge, grid-stride
__global__ void k_scatter_T(const int* __restrict__ row, const int* __restrict__ col,
                            const float* __restrict__ w, const float* __restrict__ S,
                            float* __restrict__ T, int E, int C) {
    int lane = threadIdx.x & 31;
    int wave = blockIdx.x * (blockDim.x >> 5) + (threadIdx.x >> 5);
    int nw   = gridDim.x * (blockDim.x >> 5);
    for (int e = wave; e < E; e += nw) {
        int r = row[e], c = col[e];
        float we = w[e];
        const float* Sc = S + (long)c * C;
        float*       Tr = T + (long)r * C;
        for (int f = lane; f < C; f += 32) atomAdd(&Tr[f], we * Sc[f]);
    }
}

// ---------------- pooled dense-GCN pieces ----------------
__global__ void k_deg2(const float* __restrict__ A, float* dinv2) {
    int i = blockIdx.x * blockDim.x + threadIdx.x;
    if (i >= BGR * NC) return;
    int b = i / NC, c = i % NC;
    const float* Ab = A + (long)b * NC * NC;
    float d = 1.0f;                           // +I diagonal
    for (int r = 0; r < NC; ++r) d += Ab[r * NC + c];
    dinv2[i] = (d > 0.f) ? rsqrtf(fmaxf(d, 1e-12f)) : 0.f;
}
// out[b,c,f] = dinv[c] * sum_r (A[r,c]+I) * dinv[r] * XW[b,r,f] + bias[f]
__global__ void k_dense_gcn(const float* __restrict__ A, const float* __restrict__ dinv,
                            const float* __restrict__ XW, const float* __restrict__ bias,
                            float* __restrict__ out, int F) {
    int i = blockIdx.x * blockDim.x + threadIdx.x;
    if (i >= BGR * NC * F) return;
    int f = i % F; int c = (i / F) % NC; int b = i / (NC * F);
    const float* Ab = A + (long)b * NC * NC;
    const float* db = dinv + b * NC;
    const float* Xb = XW + (long)b * NC * F;
    float acc = 0.f;
    for (int r = 0; r < NC; ++r) {
        float a = Ab[r * NC + c] + (r == c ? 1.f : 0.f);
        acc += a * db[r] * Xb[r * F + f];
    }
    out[i] = acc * db[c] + bias[f];
}

// ---------------- FC head ----------------
__global__ void k_fc(const float* __restrict__ x1cat, const float* __restrict__ x2cat,
                     const float* __restrict__ W1, const float* __restrict__ b1,
                     const float* __restrict__ W2, const float* __restrict__ b2,
                     float* __restrict__ out) {
    int b = blockIdx.x;
    __shared__ float conv[180];
    __shared__ float hid[50];
    for (int i = threadIdx.x; i < 90; i += blockDim.x) {
        conv[i]      = x1cat[b * 90 + i];
        conv[90 + i] = x2cat[b * 90 + i];
    }
    __syncthreads();
    for (int j = threadIdx.x; j < 50; j += blockDim.x) {
        float a = b1[j];
        for (int i = 0; i < 180; ++i) a += conv[i] * W1[i * 50 + j];
        hid[j] = fmaxf(a, 0.f);
    }
    __syncthreads();
    for (int j = threadIdx.x; j < 6; j += blockDim.x) {
        float a = b2[j];
        for (int i = 0; i < 50; ++i) a += hid[i] * W2[i * 6 + j];
        out[b * 6 + j] = a;
    }
}

// ---------------- host orchestration ----------------
extern "C" void kernel_launch(void* const* d_in, const int* in_sizes, int n_in,
                              void* d_out, int out_size, void* d_ws, size_t ws_size,
                              hipStream_t stream) {
    (void)in_sizes; (void)n_in; (void)out_size; (void)ws_size;
    const float* x   = (const float*)d_in[0];
    const int*   row = (const int*)d_in[1];
    const int*   col = row + EDG;
    const float* ea  = (const float*)d_in[2];

    // params: jax pytree (sorted dict keys) flatten order, starting at d_in[3]
    const float* conv_W[3]  = { (const float*)d_in[3],  (const float*)d_in[5],  (const float*)d_in[7]  };
    const float* conv_b[3]  = { (const float*)d_in[4],  (const float*)d_in[6],  (const float*)d_in[8]  };
    const float* conv2_W[3] = { (const float*)d_in[9],  (const float*)d_in[11], (const float*)d_in[13] };
    const float* conv2_b[3] = { (const float*)d_in[10], (const float*)d_in[12], (const float*)d_in[14] };
    const float* fc1W = (const float*)d_in[15]; const float* fc1b = (const float*)d_in[16];
    const float* fc2W = (const float*)d_in[17]; const float* fc2b = (const float*)d_in[18];
    const float* norm_g[3]  = { (const float*)d_in[19], (const float*)d_in[21], (const float*)d_in[23] };
    const float* norm_b[3]  = { (const float*)d_in[20], (const float*)d_in[22], (const float*)d_in[24] };
    const float* norm2_g[3] = { (const float*)d_in[25], (const float*)d_in[27], (const float*)d_in[29] };
    const float* norm2_b[3] = { (const float*)d_in[26], (const float*)d_in[28], (const float*)d_in[30] };
    const float* normp_g[3] = { (const float*)d_in[31], (const float*)d_in[33], (const float*)d_in[35] };
    const float* normp_b[3] = { (const float*)d_in[32], (const float*)d_in[34], (const float*)d_in[36] };
    const float* pconv_W[3] = { (const float*)d_in[37], (const float*)d_in[39], (const float*)d_in[41] };
    const float* pconv_b[3] = { (const float*)d_in[38], (const float*)d_in[40], (const float*)d_in[42] };
    const float* pfcW = (const float*)d_in[43]; const float* pfcb = (const float*)d_in[44];

    float* out = (float*)d_out;           // 96 logits + 1 loss

    // ---- workspace bump allocator ----
    size_t off = 0;
    auto alloc = [&](size_t nfloats) {
        float* p = (float*)((char*)d_ws + off);
        off += ((nfloats * sizeof(float) + 255) / 256) * 256;
        return p;
    };
    float* dinv_w = alloc(NODES);
    float* dinv_1 = alloc(NODES);
    float* xw     = alloc((size_t)NODES * NC);   // widest xw (100 ch)
    float* agg    = alloc((size_t)NODES * NC);
    float* hbuf   = alloc((size_t)NODES * 30);
    float* x13    = alloc((size_t)NODES * 30);
    float* scat   = alloc((size_t)NODES * 160);  // assignment-branch concat
    float* s1     = alloc((size_t)NODES * NC);   // reused later as T = A*S
    float* p      = alloc((size_t)NODES * NC);
    float* p1adj  = alloc((size_t)BGR * NC * NC);
    float* dinv2  = alloc(BGR * NC);
    float* h2     = alloc((size_t)BGR * NC * 30);
    float* xw2    = alloc((size_t)BGR * NC * 30);
    float* agg2   = alloc((size_t)BGR * NC * 30);
    float* x1cat  = alloc(BGR * 90);
    float* x2cat  = alloc(BGR * 90);
    float* mean   = alloc(128);
    float* var    = alloc(128);

    auto blocks = [](long n, int bs) { return (int)((n + bs - 1) / bs); };
    auto gemm = [&](const float* A, const float* B, float* C, int M, int N, int K,
                    int lda, int ldb, int ldc, long sA, long sB, long sC,
                    int batch, int transA) {
        int tiles = ((M + 15) / 16) * ((N + 15) / 16);
        dim3 grid((tiles + 7) / 8, batch);
        k_gemm_wmma<<<grid, 256, 0, stream>>>(A, B, C, M, N, K, lda, ldb, ldc,
                                              sA, sB, sC, transA);
    };
    const int EDGE_BLOCKS = 8192;   // grid-stride, wave-per-edge kernels

    // zero outputs (loss accumulated atomically into out[96])
    k_zero<<<1, 128, 0, stream>>>(out, 97);

    // degrees -> dinv (weighted and unweighted, self loop weight 1)
    k_fill<<<blocks(NODES, 256), 256, 0, stream>>>(dinv_w, 1.0f, NODES);
    k_fill<<<blocks(NODES, 256), 256, 0, stream>>>(dinv_1, 1.0f, NODES);
    k_edge_deg<<<blocks(EDG, 256), 256, 0, stream>>>(col, ea, dinv_w, dinv_1, EDG);
    k_dinv<<<blocks(NODES, 256), 256, 0, stream>>>(dinv_w, NODES);
    k_dinv<<<blocks(NODES, 256), 256, 0, stream>>>(dinv_1, NODES);

    // ---- level-1 weighted feature branch: 3 x (GCN + BN), track per-graph max ----
    {
        const float* hin = x; int Cin = 3; int ldin = 3;
        for (int l = 0; l < 3; ++l) {
            const int Cout = 30;
            gemm(hin, conv_W[l], xw, NODES, Cout, Cin, ldin, Cout, Cout, 0, 0, 0, 1, 0);
            k_selfloop_bias<<<blocks((long)NODES * Cout, 256), 256, 0, stream>>>(
                xw, dinv_w, conv_b[l], agg, NODES, Cout);
            k_edge_scatter<<<EDGE_BLOCKS, 256, 0, stream>>>(
                row, col, ea, dinv_w, xw, agg, EDG, Cout);
            k_bn_stats<<<Cout, 256, 0, stream>>>(agg, NODES, Cout, mean, var);
            float* outb = (l == 2) ? x13 : hbuf;
            k_bn_apply<<<blocks((long)NODES * Cout, 256), 256, 0, stream>>>(
                agg, NODES, Cout, mean, var, norm_g[l], norm_b[l], outb, Cout, Cout);
            k_colmax<<<dim3(Cout, BGR), 256, 0, stream>>>(outb, NN, Cout, x1cat, 90, l * 30);
            hin = outb; Cin = Cout; ldin = Cout;
        }
    }

    // ---- assignment branch (unweighted GCN): outputs written into scat columns ----
    {
        const float* hin = x; int Cin = 3; int ldin = 3;
        const int offs[3] = { 0, 30, 60 };
        const int couts[3] = { 30, 30, 100 };
        for (int l = 0; l < 3; ++l) {
            const int Cout = couts[l];
            gemm(hin, pconv_W[l], xw, NODES, Cout, Cin, ldin, Cout, Cout, 0, 0, 0, 1, 0);
            k_selfloop_bias<<<blocks((long)NODES * Cout, 256), 256, 0, stream>>>(
                xw, dinv_1, pconv_b[l], agg, NODES, Cout);
            k_edge_scatter<<<EDGE_BLOCKS, 256, 0, stream>>>(
                row, col, (const float*)nullptr, dinv_1, xw, agg, EDG, Cout);
            k_bn_stats<<<Cout, 256, 0, stream>>>(agg, NODES, Cout, mean, var);
            k_bn_apply<<<blocks((long)NODES * Cout, 256), 256, 0, stream>>>(
                agg, NODES, Cout, mean, var, normp_g[l], normp_b[l],
                scat + offs[l], 160, Cout);
            hin = scat + offs[l]; Cin = Cout; ldin = 160;
        }
    }

    // s1 = scat @ pool_fc.W + b  : [32768,160] x [160,100]
    gemm(scat, pfcW, s1, NODES, NC, 160, 160, NC, NC, 0, 0, 0, 1, 0);
    k_add_bias<<<blocks((long)NODES * NC, 256), 256, 0, stream>>>(s1, pfcb, NODES, NC);

    // softmax -> p ; entropy loss ; modularity loss
    k_softmax_entropy<<<blocks((long)NODES * 32, 256), 256, 0, stream>>>(s1, p, out + 96, NODES, NC);
    k_modularity<<<4096, 256, 0, stream>>>(row, col, p, out + 96, EDG, NC);

    // T = A*S via edge scatter (s1 buffer reused), then p1_adj = S^T T (batched WMMA)
    float* T = s1;
    k_zero<<<blocks((long)NODES * NC, 256), 256, 0, stream>>>(T, (long)NODES * NC);
    k_scatter_T<<<EDGE_BLOCKS, 256, 0, stream>>>(row, col, ea, p, T, EDG, NC);
    gemm(p, T, p1adj, NC, NC, NN, NC, NC, NC,
         (long)NN * NC, (long)NN * NC, (long)NC * NC, BGR, 1 /*transA*/);
    // p1_x = S^T X13 -> h2 [B,100,30]
    gemm(p, x13, h2, NC, 30, NN, NC, 30, 30,
         (long)NN * NC, (long)NN * 30, (long)NC * 30, BGR, 1 /*transA*/);

    // ---- level-2 dense GCN layers on pooled graphs ----
    k_deg2<<<blocks(BGR * NC, 256), 256, 0, stream>>>(p1adj, dinv2);
    {
        const float* hin = h2;
        for (int l = 0; l < 3; ++l) {
            gemm(hin, conv2_W[l], xw2, BGR * NC, 30, 30, 30, 30, 30, 0, 0, 0, 1, 0);
            k_dense_gcn<<<blocks(BGR * NC * 30, 256), 256, 0, stream>>>(
                p1adj, dinv2, xw2, conv2_b[l], agg2, 30);
            k_bn_stats<<<30, 256, 0, stream>>>(agg2, BGR * NC, 30, mean, var);
            k_bn_apply<<<blocks((long)BGR * NC * 30, 256), 256, 0, stream>>>(
                agg2, BGR * NC, 30, mean, var, norm2_g[l], norm2_b[l], h2, 30, 30);
            k_colmax<<<dim3(30, BGR), 256, 0, stream>>>(h2, NC, 30, x2cat, 90, l * 30);
            hin = h2;
        }
    }

    // ---- FC head: out = relu([x1|x2] @ W1 + b1) @ W2 + b2 ----
    k_fc<<<BGR, 64, 0, stream>>>(x1cat, x2cat, fc1W, fc1b, fc2W, fc2b, out);
}